// ExLoss_7017976562156
// MI455X (gfx1250) — compile-verified
//
#include <hip/hip_runtime.h>
#include <math.h>

// Problem constants (ExLoss, use_prior=False)
#define Bn 32
#define Cn 40000
#define Dn 2048
#define Tscale 10.0f
#define PM 0.2f
#define NM 0.3f

#define NTILE 256
#define KCH 32
#define KSTR (KCH + 4)   // 36 == 4 (mod 8): conflict-free WMMA frag reads
#define NCH (Dn / KCH)   // 64 K-chunks
#define NBUF 4           // LDS pipeline depth (3 chunks in flight)
#define NB ((Cn + NTILE - 1) / NTILE)   // 157 blocks over V rows

// async instructions issued per thread per chunk: 2 (A) + 8 (B) = 10
#define APC 10

// Workspace layout (floats)
#define WS_H   0           // h_loss scalar
#define WS_NTH 1           // 32: n_th_t[b] (already minus N_MARGIN)
#define WS_BU  33          // 32: per-row (lse - logit[target])
#define WS_THS 80          // NB: per-block masked-softplus sums
#define WS_THC (80 + NB)   // NB: per-block masked counts
#define WS_ACT (80 + 2*NB) // NB: per-block "found a zero row-sum" flags

typedef float v2f __attribute__((ext_vector_type(2)));
typedef float v8f __attribute__((ext_vector_type(8)));

__device__ __forceinline__ float softplusf(float x) {
  return fmaxf(x, 0.0f) + log1pf(expf(-fabsf(x)));
}

__device__ __forceinline__ float wave_sum(float v) {
#pragma unroll
  for (int m = 16; m >= 1; m >>= 1) v += __shfl_xor(v, m, 32);
  return v;
}

// CDNA5 async global->LDS copy (ASYNCcnt-tracked, no VGPR staging).
__device__ __forceinline__ void async_b128(unsigned lds_off, const float* g) {
  asm volatile("global_load_async_to_lds_b128 %0, %1, off"
               :: "v"(lds_off), "v"(g) : "memory");
}
#define WAIT_ASYNC_30() asm volatile("s_wait_asynccnt 30" ::: "memory")
#define WAIT_ASYNC_20() asm volatile("s_wait_asynccnt 20" ::: "memory")
#define WAIT_ASYNC_10() asm volatile("s_wait_asynccnt 10" ::: "memory")
#define WAIT_ASYNC_0()  asm volatile("s_wait_asynccnt 0"  ::: "memory")

// ---------------------------------------------------------------------------
// K0 (1 block, 8 waves): G = inputs@inputs.T via WMMA f32 (waves 0-3),
// n_th[b] = dot(inputs[b], V[targets[b]])/||inputs[b]|| - NM (waves 4-7),
// then h_loss from the 32x32 cosine-sim matrix.
// ---------------------------------------------------------------------------
#define K0_CH 512
#define K0_STR (K0_CH + 4)
__global__ __launch_bounds__(256) void k0_prologue(const float* __restrict__ inp,
                                                   const float* __restrict__ V,
                                                   const int* __restrict__ tg,
                                                   float* __restrict__ ws) {
  __shared__ float lA[Bn][K0_STR];
  __shared__ float gsm[Bn][Bn + 1];
  __shared__ float norm_s[Bn];
  __shared__ float dots_s[Bn];
  __shared__ int   tg_s[Bn];
  __shared__ float row4[Bn][4];

  const int t = threadIdx.x;
  const int w = t >> 5, lane = t & 31;
  const int lmod = lane & 15;
  const int khalf = (lane >> 4) * 2;

  if (t < Bn) tg_s[t] = tg[t];
  __syncthreads();

  v8f acc = {};
  float part[8];
#pragma unroll
  for (int q = 0; q < 8; ++q) part[q] = 0.f;

  for (int ch = 0; ch < Dn / K0_CH; ++ch) {
    const int kbase = ch * K0_CH;
#pragma unroll 4
    for (int i = 0; i < 64; ++i) {
      int e = t + 256 * i;
      int m = e >> 9, k = e & 511;
      lA[m][k] = inp[(size_t)m * Dn + kbase + k];
    }
    __syncthreads();
    if (w < 4) {
      const int ti = (w >> 1) * 16, tj = (w & 1) * 16;
      for (int k0 = 0; k0 < K0_CH; k0 += 4) {
        v2f a, b;
        a.x = lA[ti + lmod][k0 + khalf];
        a.y = lA[ti + lmod][k0 + khalf + 1];
        b.x = lA[tj + lmod][k0 + khalf];
        b.y = lA[tj + lmod][k0 + khalf + 1];
        acc = __builtin_amdgcn_wmma_f32_16x16x4_f32(false, a, false, b,
                                                    (short)0, acc, false, false);
      }
    } else {
      const int wq = w - 4;
#pragma unroll
      for (int q = 0; q < 8; ++q) {
        const int bb = wq * 8 + q;
        const float* vr = V + (size_t)tg_s[bb] * Dn + kbase;
        float s = 0.f;
        for (int k = lane; k < K0_CH; k += 32) s += lA[bb][k] * vr[k];
        part[q] += s;
      }
    }
    __syncthreads();
  }

  if (w < 4) {
    const int ti = (w >> 1) * 16, tj = (w & 1) * 16;
    const int mrow = ti + (lane >> 4) * 8;
#pragma unroll
    for (int r = 0; r < 8; ++r) gsm[mrow + r][tj + lmod] = acc[r];
  } else {
    const int wq = w - 4;
#pragma unroll
    for (int q = 0; q < 8; ++q) {
      float s = wave_sum(part[q]);
      if (lane == 0) dots_s[wq * 8 + q] = s;
    }
  }
  __syncthreads();
  if (t < Bn) norm_s[t] = sqrtf(gsm[t][t]);
  __syncthreads();
  if (t < Bn) {
    ws[WS_NTH + t] = dots_s[t] / norm_s[t] - NM;
    const int i = t;
    float pmax = -2.f, nmin = 2.f;
    for (int j = 0; j < Bn; ++j) {
      bool pos = (tg_s[i] == tg_s[j]) && (i != j);
      float s = gsm[i][j] / (norm_s[i] * norm_s[j]);
      if (pos) { pmax = fmaxf(pmax, s); nmin = fminf(nmin, s); }
    }
    const float pthr = pmax - PM, nthr = nmin - NM;
    float hps = 0.f, hpc = 0.f, hns = 0.f, hnc = 0.f;
    for (int j = 0; j < Bn; ++j) {
      bool same = (tg_s[i] == tg_s[j]);
      bool pos = same && (i != j);
      float s = gsm[i][j] / (norm_s[i] * norm_s[j]);
      if (pos && s < pthr)  { hps += softplusf(-s); hpc += 1.f; }
      if (!same && s > nthr) { hns += softplusf(s);  hnc += 1.f; }
    }
    row4[i][0] = hps; row4[i][1] = hpc; row4[i][2] = hns; row4[i][3] = hnc;
  }
  __syncthreads();
  if (t == 0) {
    float hps = 0.f, hpc = 0.f, hns = 0.f, hnc = 0.f;
    for (int i = 0; i < Bn; ++i) {
      hps += row4[i][0]; hpc += row4[i][1];
      hns += row4[i][2]; hnc += row4[i][3];
    }
    ws[WS_H] = (hpc > 0.f ? hps / hpc : 0.f) + (hnc > 0.f ? hns / hnc : 0.f);
  }
}

// ---------------------------------------------------------------------------
// K1: single fused pass over V with a depth-3 async global->LDS pipeline.
// Each block owns 256 V rows; each wave owns two 16-col N strips
// (8 accumulators), so 4 A-frags feed 8 WMMAs per K-step.
// A = [inputs(32) ; V[targets](32)], B = V tile. Rows 0-31 of D -> outputs*T.
// Rows 32-63 -> nsims_t masked-BCE partials. V row sums -> 'active' check.
// ---------------------------------------------------------------------------
__global__ __launch_bounds__(256) void k1_main(const float* __restrict__ inp,
                                               const float* __restrict__ V,
                                               const int* __restrict__ tg,
                                               const float* __restrict__ nth,
                                               float* __restrict__ outp,  // d_out+1
                                               float* __restrict__ ws) {
  __shared__ float lA[NBUF][64][KSTR];      // 36.9 KB
  __shared__ float lB[NBUF][NTILE][KSTR];   // 147.5 KB
  __shared__ int   tg_s[Bn];
  __shared__ float nth_s[Bn];
  __shared__ float red[256];
  __shared__ float redc[256];
  __shared__ int   act_s;

  const int t = threadIdx.x;
  const int w = t >> 5, lane = t & 31;
  const int lmod = lane & 15;
  const int khalf = (lane >> 4) * 2;
  const int c0 = blockIdx.x * NTILE;
  const int nw = w * 32;  // this wave's 32-column region (two 16-col strips)

  if (t < Bn) { tg_s[t] = tg[t]; nth_s[t] = nth[t]; }
  if (t == 0) act_s = 0;
  __syncthreads();

  // LDS byte offsets of the staging buffers (flat LDS addr keeps offset in [31:0])
  const unsigned lA_base = (unsigned)(uintptr_t)&lA[0][0][0];
  const unsigned lB_base = (unsigned)(uintptr_t)&lB[0][0][0];

  // Per-thread async-issue geometry (16B granules, 128B-contiguous per 8 lanes)
  const int jA = t & 7;          // column granule within a 32-float row
  const int mA0 = t >> 3;        // A row for i=0 (i=1 -> +32)
  const int nB0 = t >> 3;        // B row for i=0 (i -> +32*i)

  // issue one chunk's async loads (10 instructions per thread)
  auto issue_chunk = [&](int ck) {
    const int buf = ck & (NBUF - 1);
    const int kbase = ck * KCH;
    const int col = 4 * jA;
    // A: 2 granules
#pragma unroll
    for (int i = 0; i < 2; ++i) {
      int m = mA0 + 32 * i;
      const float* g = (m < Bn)
          ? inp + (size_t)m * Dn + kbase + col
          : V + (size_t)tg_s[m - Bn] * Dn + kbase + col;
      unsigned lds = lA_base + (unsigned)(((buf * 64 + m) * KSTR + col) * 4);
      async_b128(lds, g);
    }
    // B: 8 granules
#pragma unroll
    for (int i = 0; i < 8; ++i) {
      int n = nB0 + 32 * i;
      int c = c0 + n;
      int crow = c < Cn ? c : Cn - 1;
      const float* g = V + (size_t)crow * Dn + kbase + col;
      unsigned lds = lB_base + (unsigned)(((buf * NTILE + n) * KSTR + col) * 4);
      async_b128(lds, g);
    }
  };

  v8f acc[4][2] = {};
  float rs = 0.f;  // V-row partial sum (one row per thread)

  // prime the pipeline: chunks 0,1,2 in flight
  issue_chunk(0);
  issue_chunk(1);
  issue_chunk(2);

  for (int kb = 0; kb < NCH; ++kb) {
    if (kb + 3 < NCH) issue_chunk(kb + 3);
    // wait until this wave's chunk-kb loads have landed (in-order completion)
    const int rem = NCH - 1 - kb;
    if (rem >= 3)      WAIT_ASYNC_30();
    else if (rem == 2) WAIT_ASYNC_20();
    else if (rem == 1) WAIT_ASYNC_10();
    else               WAIT_ASYNC_0();
    __syncthreads();   // everyone's chunk-kb data visible

    const int cur = kb & (NBUF - 1);

    // row-sum for 'active' check (one V row per thread)
    {
      float s = 0.f;
#pragma unroll 8
      for (int k = 0; k < KCH; ++k) s += lB[cur][t][k];
      rs += s;
    }

    // WMMA: 8 K-steps x (4 M-blocks x 2 N-strips) per chunk, per wave
    for (int k0 = 0; k0 < KCH; k0 += 4) {
      v2f b0, b1;
      b0.x = lB[cur][nw + lmod][k0 + khalf];
      b0.y = lB[cur][nw + lmod][k0 + khalf + 1];
      b1.x = lB[cur][nw + 16 + lmod][k0 + khalf];
      b1.y = lB[cur][nw + 16 + lmod][k0 + khalf + 1];
#pragma unroll
      for (int mb = 0; mb < 4; ++mb) {
        v2f a;
        a.x = lA[cur][mb * 16 + lmod][k0 + khalf];
        a.y = lA[cur][mb * 16 + lmod][k0 + khalf + 1];
        acc[mb][0] = __builtin_amdgcn_wmma_f32_16x16x4_f32(false, a, false, b0,
                                                           (short)0, acc[mb][0],
                                                           false, false);
        acc[mb][1] = __builtin_amdgcn_wmma_f32_16x16x4_f32(false, a, false, b1,
                                                           (short)0, acc[mb][1],
                                                           false, false);
      }
    }
    __syncthreads();   // buffer 'cur' free for the issue in the next iteration
  }

  {
    int c = c0 + t;
    if (c < Cn && rs == 0.0f) atomicOr(&act_s, 1);
  }

  // Epilogue. D layout: VGPR r -> M = (lane<16 ? r : 8+r), N = lane&15.
  const int mhalf = (lane >> 4) * 8;
  float thsum = 0.f, thcnt = 0.f;
#pragma unroll
  for (int s = 0; s < 2; ++s) {
    const int n = c0 + nw + s * 16 + lmod;
#pragma unroll
    for (int mb = 0; mb < 2; ++mb) {
#pragma unroll
      for (int r = 0; r < 8; ++r) {
        int m = mb * 16 + mhalf + r;
        if (n < Cn) outp[(size_t)m * Cn + n] = acc[mb][s][r] * Tscale;
      }
    }
#pragma unroll
    for (int mb = 2; mb < 4; ++mb) {
#pragma unroll
      for (int r = 0; r < 8; ++r) {
        int bb = (mb - 2) * 16 + mhalf + r;
        float v = acc[mb][s][r];
        if (n < Cn && v < 0.9999f && v > nth_s[bb]) {
          thsum += softplusf(v);
          thcnt += 1.f;
        }
      }
    }
  }
  red[t] = thsum; redc[t] = thcnt;
  __syncthreads();
  for (int off = 128; off >= 1; off >>= 1) {
    if (t < off) { red[t] += red[t + off]; redc[t] += redc[t + off]; }
    __syncthreads();
  }
  if (t == 0) {
    ws[WS_THS + blockIdx.x] = red[0];
    ws[WS_THC + blockIdx.x] = redc[0];
    ws[WS_ACT + blockIdx.x] = (float)act_s;
  }
}

// ---------------------------------------------------------------------------
// K2: per-row logsumexp over outputs (32 blocks); stores lse - logit[target].
// ---------------------------------------------------------------------------
__global__ __launch_bounds__(256) void k2_rowred(const float* __restrict__ outp,
                                                 const int* __restrict__ tg,
                                                 float* __restrict__ ws) {
  __shared__ float red[256];
  const int b = blockIdx.x;
  const int t = threadIdx.x;
  const float* row = outp + (size_t)b * Cn;

  float mx = -3.4e38f;
  for (int c = t; c < Cn; c += 256) mx = fmaxf(mx, row[c]);
  red[t] = mx;
  __syncthreads();
  for (int off = 128; off >= 1; off >>= 1) {
    if (t < off) red[t] = fmaxf(red[t], red[t + off]);
    __syncthreads();
  }
  mx = red[0];
  __syncthreads();

  float se = 0.f;
  for (int c = t; c < Cn; c += 256) se += expf(row[c] - mx);
  red[t] = se;
  __syncthreads();
  for (int off = 128; off >= 1; off >>= 1) {
    if (t < off) red[t] += red[t + off];
    __syncthreads();
  }
  if (t == 0) {
    float lse = mx + logf(red[0]);
    ws[WS_BU + b] = lse - row[tg[b]];
  }
}

// ---------------------------------------------------------------------------
// K3: fixed-order final combine -> loss scalar at d_out[0].
// ---------------------------------------------------------------------------
__global__ void k3_final(const float* __restrict__ ws, float* __restrict__ out0) {
  if (threadIdx.x == 0 && blockIdx.x == 0) {
    float bu = 0.f;
    for (int b = 0; b < Bn; ++b) bu += ws[WS_BU + b];
    bu /= (float)Bn;
    float ths = 0.f, thc = 0.f; int bad = 0;
    for (int i = 0; i < NB; ++i) {
      ths += ws[WS_THS + i];
      thc += ws[WS_THC + i];
      if (ws[WS_ACT + i] != 0.f) bad = 1;
    }
    float th = (thc > 0.f) ? ths / thc : 0.f;
    if (bad) th = 0.f;  // active == false -> th_loss = 0
    out0[0] = 1.0f * bu + 1.0f * ws[WS_H] + 3.0f * th;
  }
}

extern "C" void kernel_launch(void* const* d_in, const int* in_sizes, int n_in,
                              void* d_out, int out_size, void* d_ws, size_t ws_size,
                              hipStream_t stream) {
  (void)in_sizes; (void)n_in; (void)out_size; (void)ws_size;
  const float* inp = (const float*)d_in[0];
  const float* V   = (const float*)d_in[1];
  const int*   tg  = (const int*)d_in[2];
  float* out = (float*)d_out;   // [0]=loss, [1..]=outputs (32 x 40000)
  float* ws  = (float*)d_ws;

  k0_prologue<<<1, 256, 0, stream>>>(inp, V, tg, ws);
  k1_main<<<NB, 256, 0, stream>>>(inp, V, tg, ws + WS_NTH, out + 1, ws);
  k2_rowred<<<Bn, 256, 0, stream>>>(out + 1, tg, ws);
  k3_final<<<1, 32, 0, stream>>>(ws, out);
}